// NodeAttention_64493228917344
// MI455X (gfx1250) — compile-verified
//
#include <hip/hip_runtime.h>

#define B_   8
#define N_   512
#define D_   768
#define DEP_ 64

typedef __attribute__((ext_vector_type(2))) float v2f;
typedef __attribute__((ext_vector_type(8))) float v8f;

// ---------------- Kernel 1: per-node projections s_nbr, s_asp ----------------
__global__ __launch_bounds__(256) void node_scores_kernel(
    const float* __restrict__ features, const float* __restrict__ w_att,
    float* __restrict__ s_nbr, float* __restrict__ s_asp) {
  const int node = blockIdx.x;                 // b*N + j
  const float* f  = features + (size_t)node * D_;
  const float* wn = w_att;                     // w_nbr = w_att[0:D)
  const float* wa = w_att + D_ + DEP_;         // w_asp = w_att[D+DEP:2D+DEP)
  const int tid = threadIdx.x;
  float pn = 0.f, pa = 0.f;
  for (int d = tid; d < D_; d += 256) {
    const float fv = f[d];
    pn += fv * wn[d];
    pa += fv * wa[d];
  }
  __shared__ float rn[256], ra[256];
  rn[tid] = pn; ra[tid] = pa;
  __syncthreads();
  for (int s = 128; s > 0; s >>= 1) {
    if (tid < s) { rn[tid] += rn[tid + s]; ra[tid] += ra[tid + s]; }
    __syncthreads();
  }
  if (tid == 0) { s_nbr[node] = rn[0]; s_asp[node] = ra[0]; }
}

// ------- Kernel 2: stream adj (537MB), masked scores + row softmax ----------
__global__ __launch_bounds__(256) void edge_softmax_kernel(
    const float* __restrict__ adj, const float* __restrict__ w_att,
    const int* __restrict__ aspect, const float* __restrict__ s_nbr,
    const float* __restrict__ s_asp, float* __restrict__ weights,
    int* __restrict__ uflag) {
  const int row = blockIdx.x;                  // b*N + i
  const int b   = row >> 9;                    // row / N_
  const int tid = threadIdx.x;
  const int g   = tid >> 4;                    // group 0..15 (one j per group)
  const int l   = tid & 15;                    // lane within group

  __shared__ float snb[N_];
  __shared__ float sc[N_];
  __shared__ float red[8];

  snb[tid]       = s_nbr[b * N_ + tid];
  snb[tid + 256] = s_nbr[b * N_ + tid + 256];
  const float sasp = s_asp[row];
  const float4 w4 = ((const float4*)(w_att + D_))[l];   // w_dep slice
  __syncthreads();

  const float* arow = adj + (size_t)row * N_ * DEP_;
  for (int jt = 0; jt < N_ / 16; ++jt) {
    const int j = jt * 16 + g;
    const float4 a = ((const float4*)(arow + (size_t)j * DEP_))[l];
    float p = a.x * w4.x + a.y * w4.y + a.z * w4.z + a.w * w4.w;
    int nz = (a.x != 0.f) | (a.y != 0.f) | (a.z != 0.f) | (a.w != 0.f);
    #pragma unroll
    for (int off = 8; off >= 1; off >>= 1) {
      p  += __shfl_xor(p, off, 16);
      nz |= __shfl_xor(nz, off, 16);
    }
    if (l == 0) sc[j] = nz ? (snb[j] + p + sasp) : -1e30f;
  }
  __syncthreads();

  // block max over 512 scores
  float m = fmaxf(sc[tid], sc[tid + 256]);
  #pragma unroll
  for (int off = 16; off >= 1; off >>= 1) m = fmaxf(m, __shfl_xor(m, off, 32));
  if ((tid & 31) == 0) red[tid >> 5] = m;
  __syncthreads();
  m = red[0];
  #pragma unroll
  for (int w = 1; w < 8; ++w) m = fmaxf(m, red[w]);
  __syncthreads();

  // sum of exp (masked entries: exp(-1e30 - m) underflows to exact 0)
  const float e0 = __expf(sc[tid] - m);
  const float e1 = __expf(sc[tid + 256] - m);
  float s = e0 + e1;
  #pragma unroll
  for (int off = 16; off >= 1; off >>= 1) s += __shfl_xor(s, off, 32);
  if ((tid & 31) == 0) red[tid >> 5] = s;
  __syncthreads();
  s = 0.f;
  #pragma unroll
  for (int w = 0; w < 8; ++w) s += red[w];
  const float inv = 1.f / s;

  float* wr = weights + (size_t)row * N_;
  wr[tid]       = e0 * inv;
  wr[tid + 256] = e1 * inv;
  if (tid == 0)
    uflag[row] = (aspect[row] != 0) && (m > -1e29f);
}

// ------ Kernel 3: agg = weights @ features via V_WMMA_F32_16X16X4_F32 -------
// One wave per 16x64 output tile (4 n-subtiles reuse the A fragment).
__global__ __launch_bounds__(256) void agg_wmma_kernel(
    const float* __restrict__ weights, const float* __restrict__ features,
    const int* __restrict__ uflag, float* __restrict__ out) {
  const int wave = threadIdx.x >> 5;
  const int lane = threadIdx.x & 31;
  const int tile = blockIdx.x * 8 + wave;      // tiles/batch = 32*12 = 384
  const int b  = tile / 384;
  const int r  = tile % 384;
  const int i0 = (r / 12) * 16;
  const int d0 = (r % 12) * 64;

  const float* Wb = weights  + (size_t)b * N_ * N_;
  const float* Fb = features + (size_t)b * N_ * D_;

  const int half = lane >> 4;                  // 0: K0/K1, 1: K2/K3
  const int l    = lane & 15;

  v8f acc[4];
  const v8f vzero = {0.f, 0.f, 0.f, 0.f, 0.f, 0.f, 0.f, 0.f};
  #pragma unroll
  for (int s = 0; s < 4; ++s) acc[s] = vzero;

  for (int k0 = 0; k0 < N_; k0 += 4) {
    const int kb = k0 + half * 2;
    // A fragment: W[i0+l, kb], W[i0+l, kb+1]
    const float* ap = Wb + (size_t)(i0 + l) * N_ + kb;
    v2f a; a.x = ap[0]; a.y = ap[1];
    // B fragments: F[kb, d0+sub*16+l], F[kb+1, ...]
    const float* bp0 = Fb + (size_t)kb * D_ + d0 + l;
    const float* bp1 = bp0 + D_;
    #pragma unroll
    for (int sub = 0; sub < 4; ++sub) {
      v2f bv; bv.x = bp0[sub * 16]; bv.y = bp1[sub * 16];
      acc[sub] = __builtin_amdgcn_wmma_f32_16x16x4_f32(
          /*neg_a=*/false, a, /*neg_b=*/false, bv,
          /*c_mod=*/(short)0, acc[sub], /*reuse_a=*/false, /*reuse_b=*/false);
    }
  }

  // Store with fused row-select: out = u ? agg : features
  #pragma unroll
  for (int rr = 0; rr < 8; ++rr) {
    const int rowi = i0 + rr + half * 8;       // C/D layout: lanes16-31 -> M+8
    const int u = uflag[b * N_ + rowi];
    const size_t base = ((size_t)b * N_ + rowi) * D_;
    #pragma unroll
    for (int sub = 0; sub < 4; ++sub) {
      const int col = d0 + sub * 16 + l;
      const float val = acc[sub][rr];
      out[base + col] = u ? val : features[base + col];
    }
  }
}

extern "C" void kernel_launch(void* const* d_in, const int* in_sizes, int n_in,
                              void* d_out, int out_size, void* d_ws, size_t ws_size,
                              hipStream_t stream) {
  const float* features = (const float*)d_in[0];   // [B,N,D] f32
  const int*   aspect   = (const int*)  d_in[1];   // [B,N]   i32
  const float* adj      = (const float*)d_in[2];   // [B,N,N,DEP] f32
  const float* w_att    = (const float*)d_in[3];   // [2D+DEP] f32
  float* out = (float*)d_out;

  float* ws      = (float*)d_ws;
  float* s_nbr   = ws;                             // B*N floats
  float* s_asp   = ws + B_ * N_;                   // B*N floats
  int*   uflag   = (int*)(ws + 2 * B_ * N_);       // B*N ints
  float* weights = ws + 3 * B_ * N_;               // B*N*N floats (8 MB)

  node_scores_kernel<<<B_ * N_, 256, 0, stream>>>(features, w_att, s_nbr, s_asp);
  edge_softmax_kernel<<<B_ * N_, 256, 0, stream>>>(adj, w_att, aspect, s_nbr,
                                                   s_asp, weights, uflag);
  agg_wmma_kernel<<<(B_ * 384) / 8, 256, 0, stream>>>(weights, features, uflag, out);
}